// GCN_graph_56178172232068
// MI455X (gfx1250) — compile-verified
//
#include <hip/hip_runtime.h>
#include <hip/hip_bf16.h>

// ---------------------------------------------------------------------------
// GCN forward for MI455X (gfx1250, wave32, WMMA).
//   N=50000 nodes, E=800000 edges, F=H=128, 3 layers, 512 graphs, 10 classes.
// Memory-bound on edge scatter (~2.6 GB, L2-resident); GEMMs done with
// V_WMMA_F32_16X16X4_F32 (fp32, matches reference precision).
// ---------------------------------------------------------------------------

#define N_NODES   50000
#define N_EDGES   800000
#define N_FEAT    128
#define N_LAYER   3
#define N_GRAPHS  512
#define N_CLASS   10

typedef __attribute__((ext_vector_type(2))) float v2f;
typedef __attribute__((ext_vector_type(8))) float v8f;

// ---------------------------------------------------------------- degree ----
__global__ void gcn_deg_init(float* __restrict__ deg) {
    int i = blockIdx.x * blockDim.x + threadIdx.x;
    if (i < N_NODES) deg[i] = 1.0f;            // self-loop contributes 1
}

__global__ void gcn_deg_count(const int* __restrict__ ei, float* __restrict__ deg) {
    int e = blockIdx.x * blockDim.x + threadIdx.x;
    if (e < N_EDGES) atomicAdd(&deg[ei[N_EDGES + e]], 1.0f);   // col = target
}

__global__ void gcn_deg_rsqrt(float* __restrict__ deg) {
    int i = blockIdx.x * blockDim.x + threadIdx.x;
    if (i < N_NODES) deg[i] = rsqrtf(deg[i]);  // deg >= 1 always (self-loops)
}

// ------------------------------------------------------------- WMMA GEMM ----
// hw = A @ W  (A: n x 128 fp32, W: 128 x 128), fused epilogue:
//   agg = dinv^2 * hw   (self-loop term of the normalized aggregation)
// Block = 256 threads = 8 waves. Block handles 16 rows; wave w handles the
// 16-column slab [16w, 16w+16). A-tile (16x128 = 8 KB) staged in LDS once.
__global__ void __launch_bounds__(256)
gcn_gemm_wmma(const float* __restrict__ A, const float* __restrict__ W,
              const float* __restrict__ dinv, float* __restrict__ hw,
              float* __restrict__ agg) {
    __shared__ float lds_a[16 * 128];
    const int row0 = blockIdx.x * 16;

    // Stage A tile: coalesced, 2048 floats / 256 threads = 8 each.
    for (int i = threadIdx.x; i < 16 * 128; i += 256) {
        int r = i >> 7, c = i & 127;
        lds_a[i] = A[(size_t)(row0 + r) * 128 + c];
    }
    __syncthreads();

    const int lane = threadIdx.x & 31;
    const int wave = threadIdx.x >> 5;          // 0..7 -> N tile
    const int m  = lane & 15;                   // A: M = lane&15
    const int kb = (lane >> 4) * 2;             // A/B: K half select
    const int n  = (wave << 4) + (lane & 15);   // global output column

    v8f c = {};
    #pragma unroll
    for (int k = 0; k < 128; k += 4) {
        v2f a, b;
        a.x = lds_a[m * 128 + k + kb];
        a.y = lds_a[m * 128 + k + kb + 1];
        b.x = W[(size_t)(k + kb) * 128 + n];
        b.y = W[(size_t)(k + kb + 1) * 128 + n];
        // D = A(16x4) * B(4x16) + C  -> v_wmma_f32_16x16x4_f32
        c = __builtin_amdgcn_wmma_f32_16x16x4_f32(
                false, a, false, b, (short)0, c, false, false);
    }

    // C/D layout: VGPR r holds M = r + 8*(lane>=16), N = lane&15.
    const int mbase = row0 + ((lane >> 4) << 3);
    #pragma unroll
    for (int r = 0; r < 8; ++r) {
        int gm = mbase + r;
        float v = c[r];
        float d = dinv[gm];
        hw [(size_t)gm * 128 + n] = v;
        agg[(size_t)gm * 128 + n] = d * d * v;   // self-loop init
    }
}

// ------------------------------------------------------- edge aggregation ---
// One wave per edge: lane reads float4 (b128) of hw[src], scales by
// dinv[src]*dinv[dst], atomically adds into agg[dst]. L2-resident scatter.
__global__ void __launch_bounds__(256)
gcn_edge_scatter(const int* __restrict__ ei, const float* __restrict__ dinv,
                 const float* __restrict__ hw, float* __restrict__ agg) {
    const int lane   = threadIdx.x & 31;
    const int wid    = (blockIdx.x * blockDim.x + threadIdx.x) >> 5;
    const int nwaves = (gridDim.x * blockDim.x) >> 5;
    for (int e = wid; e < N_EDGES; e += nwaves) {
        const int s = ei[e];                 // row = source j
        const int t = ei[N_EDGES + e];       // col = target i
        const float w = dinv[s] * dinv[t];
        const float4 v = ((const float4*)(hw + (size_t)s * 128))[lane];
        float* dst = agg + (size_t)t * 128 + lane * 4;
        atomicAdd(dst + 0, v.x * w);
        atomicAdd(dst + 1, v.y * w);
        atomicAdd(dst + 2, v.z * w);
        atomicAdd(dst + 3, v.w * w);
    }
}

// ------------------------------------------------------------ bias + ReLU ---
__global__ void gcn_bias_relu(const float* __restrict__ agg,
                              const float* __restrict__ bias,
                              float* __restrict__ h) {
    int i = blockIdx.x * blockDim.x + threadIdx.x;
    if (i < N_NODES * N_FEAT) {
        float v = agg[i] + bias[i & 127];
        h[i] = v > 0.0f ? v : 0.0f;
    }
}

// ---------------------------------------------------------------- pooling ---
__global__ void gcn_zero(float* __restrict__ p, int n) {
    int i = blockIdx.x * blockDim.x + threadIdx.x;
    if (i < n) p[i] = 0.0f;
}

__global__ void gcn_pool(const float* __restrict__ h, const int* __restrict__ batch,
                         float* __restrict__ pooled) {
    int i = blockIdx.x * blockDim.x + threadIdx.x;
    if (i < N_NODES * N_FEAT) {
        int node = i >> 7, f = i & 127;
        atomicAdd(&pooled[(size_t)batch[node] * 128 + f], h[i]);
    }
}

// ----------------------------------------------------------- output layer ---
__global__ void gcn_out_gemm(const float* __restrict__ pooled,
                             const float* __restrict__ Wout,
                             const float* __restrict__ bout,
                             float* __restrict__ out) {
    int i = blockIdx.x * blockDim.x + threadIdx.x;   // g * 10 + c
    if (i < N_GRAPHS * N_CLASS) {
        int g = i / N_CLASS, cc = i % N_CLASS;
        float acc = bout[cc];
        #pragma unroll 4
        for (int k = 0; k < 128; ++k)
            acc += pooled[(size_t)g * 128 + k] * Wout[(size_t)k * N_CLASS + cc];
        out[i] = acc;
    }
}

// ------------------------------------------------------------------ driver --
extern "C" void kernel_launch(void* const* d_in, const int* in_sizes, int n_in,
                              void* d_out, int out_size, void* d_ws, size_t ws_size,
                              hipStream_t stream) {
    const float* x     = (const float*)d_in[0];        // [N, 128]
    const int*   ei    = (const int*)  d_in[1];        // [2, E]
    const int*   batch = (const int*)  d_in[2];        // [N]
    const float* W     = (const float*)d_in[3];        // [3, 128, 128]
    const float* b     = (const float*)d_in[4];        // [3, 128]
    const float* Wout  = (const float*)d_in[5];        // [128, 10]
    const float* bout  = (const float*)d_in[6];        // [10]
    float* out = (float*)d_out;                        // [512, 10]

    // Workspace partition (all offsets 1 MB aligned; total ~80 MB).
    char* ws = (char*)d_ws;
    const size_t FEAT_BYTES = (size_t)N_NODES * N_FEAT * 4;   // 25.6 MB
    float* dinv   = (float*)(ws);                              // 200 KB
    float* hw     = (float*)(ws + (1u << 20));
    float* agg    = (float*)(ws + (1u << 20) + 0x1900000);     // +26 MB
    float* h      = (float*)(ws + (1u << 20) + 2 * 0x1900000);
    float* pooled = (float*)(ws + (1u << 20) + 3 * 0x1900000);
    (void)FEAT_BYTES; (void)ws_size; (void)in_sizes; (void)n_in; (void)out_size;

    const int T = 256;

    // 1) Symmetric GCN normalization: dinv = rsqrt(1 + in-degree)
    gcn_deg_init <<<(N_NODES + T - 1) / T, T, 0, stream>>>(dinv);
    gcn_deg_count<<<(N_EDGES + T - 1) / T, T, 0, stream>>>(ei, dinv);
    gcn_deg_rsqrt<<<(N_NODES + T - 1) / T, T, 0, stream>>>(dinv);

    // 2) Three GCNConv layers
    const float* layer_in = x;
    for (int l = 0; l < N_LAYER; ++l) {
        const float* Wl = W + (size_t)l * 128 * 128;
        const float* bl = b + (size_t)l * 128;
        gcn_gemm_wmma<<<N_NODES / 16, 256, 0, stream>>>(layer_in, Wl, dinv, hw, agg);
        gcn_edge_scatter<<<8192, 256, 0, stream>>>(ei, dinv, hw, agg);
        gcn_bias_relu<<<(N_NODES * N_FEAT + T - 1) / T, T, 0, stream>>>(agg, bl, h);
        layer_in = h;
    }

    // 3) global_add_pool + output projection
    gcn_zero<<<(N_GRAPHS * N_FEAT + T - 1) / T, T, 0, stream>>>(pooled, N_GRAPHS * N_FEAT);
    gcn_pool<<<(N_NODES * N_FEAT + T - 1) / T, T, 0, stream>>>(h, batch, pooled);
    gcn_out_gemm<<<(N_GRAPHS * N_CLASS + T - 1) / T, T, 0, stream>>>(pooled, Wout, bout, out);
}